// SurfaceAbstractionCN2_67997922230561
// MI455X (gfx1250) — compile-verified
//
#include <hip/hip_runtime.h>

typedef __attribute__((ext_vector_type(2))) float v2f;
typedef __attribute__((ext_vector_type(8))) float v8f;

#define NPTS     32768
#define NSAMP    32
#define CFEAT    64
#define ROWS_TOT (NPTS * NSAMP)
#define XS       76          // LDS row stride (floats), bank-conflict-free for 16-row frag reads
#define WAVES    8
#define TPB      (WAVES * 32)
#define NWG      512         // 512*8 = 4096 waves -> exactly 8 points per wave

// workspace layout (floats)
#define WS_SUML 0
#define WS_SQL  64
#define WS_SUMF 128
#define WS_SQF  192
#define WS_SUM1 256
#define WS_SQ1  320
#define WS_SUM2 384
#define WS_SQ2  512
#define WS_AL   640
#define WS_AF   704
#define WS_DC   768
#define WS_A1   832
#define WS_D1   896
#define WS_A2   960
#define WS_D2   1088
#define WS_NACC 640          // accumulator region that must be zeroed each launch

__device__ __forceinline__ v8f wmma_f32(v2f a, v2f b, v8f c) {
  // D = A(16x4, f32) * B(4x16, f32) + C(16x16, f32)
  return __builtin_amdgcn_wmma_f32_16x16x4_f32(false, a, false, b, (short)0, c, false, false);
}

// Combined stage-A weight matrix W^T[k][c] for X layout:
// cols 0-2: delta-center, 3: zero, 4-6: normal, 7: zero, 8-71: feature
__device__ __forceinline__ float wA_elem(const float* __restrict__ wl,
                                         const float* __restrict__ wf,
                                         int k, int c) {
  if (k < 3)            return wl[c * 3 + k];        // w_l0 [64,3]
  if (k == 3 || k == 7) return 0.0f;                 // padding
  if (k < 7)            return wf[c * 67 + (k - 4)]; // w_f0 cols 0..2 (normal)
  return wf[c * 67 + (k - 5)];                       // w_f0 cols 3..66 (feature)
}

template<int STAGE>
__global__ __launch_bounds__(TPB) void sa_pass(
    const float* __restrict__ center, const float* __restrict__ normal,
    const float* __restrict__ feature, const int* __restrict__ gidx,
    const float* __restrict__ w_l0, const float* __restrict__ w_f0,
    const float* __restrict__ w1,  const float* __restrict__ w2,
    float* __restrict__ ws, float* __restrict__ out_feat)
{
  __shared__ float xbuf[WAVES][16 * XS];
  __shared__ float lstats[256];

  const int tid  = threadIdx.x;
  const int lane = tid & 31;
  const int wave = tid >> 5;
  const int half = lane >> 4;   // 0: rows 0..7 of C/D, K 0..1 of frags; 1: rows 8..15, K 2..3
  const int r    = lane & 15;   // row (A) / column (B,C,D) within tile

  if (STAGE < 4) {
    for (int i = tid; i < 256; i += TPB) lstats[i] = 0.0f;
  }
  __syncthreads();

  float* xb = xbuf[wave];

  // hoisted per-lane BN scale/shift (channel = r + 16*c)
  float aL[4], aF[4], dC[4], a1v[4], d1v[4], a2v[8], d2v[8];
  if (STAGE >= 2) {
    #pragma unroll
    for (int c = 0; c < 4; ++c) {
      int ch = r + 16 * c;
      aL[c] = ws[WS_AL + ch]; aF[c] = ws[WS_AF + ch]; dC[c] = ws[WS_DC + ch];
    }
  }
  if (STAGE >= 3) {
    #pragma unroll
    for (int c = 0; c < 4; ++c) { int ch = r + 16 * c; a1v[c] = ws[WS_A1 + ch]; d1v[c] = ws[WS_D1 + ch]; }
  }
  if (STAGE >= 4) {
    #pragma unroll
    for (int c = 0; c < 8; ++c) { int ch = r + 16 * c; a2v[c] = ws[WS_A2 + ch]; d2v[c] = ws[WS_D2 + ch]; }
  }

  // per-lane running stat accumulators
  float sA[8] = {0,0,0,0,0,0,0,0}, qA[8] = {0,0,0,0,0,0,0,0};
  float s1[4] = {0,0,0,0},         q1[4] = {0,0,0,0};
  float s2[8] = {0,0,0,0,0,0,0,0}, q2[8] = {0,0,0,0,0,0,0,0};

  const int gwave  = blockIdx.x * WAVES + wave;
  const int nwaves = gridDim.x * WAVES;
  const v8f vzero  = {0.f,0.f,0.f,0.f,0.f,0.f,0.f,0.f};

  for (int m = gwave; m < NPTS; m += nwaves) {
    float omax[8];
    if (STAGE == 4) {
      #pragma unroll
      for (int c = 0; c < 8; ++c) omax[c] = -3.402823466e38f;
    }

    for (int t = 0; t < 2; ++t) {          // two 16-row tiles per point (32 neighbors)
      // ---------- stage gathered X tile into LDS ----------
      const int s = t * 16 + r;
      const int j = gidx[m * NSAMP + s];
      const float* fj = feature + (size_t)j * CFEAT;
      if (half == 0) {
        const float4* f4 = (const float4*)fj;
        #pragma unroll
        for (int q = 0; q < 8; ++q) {
          float4 v = f4[q];
          float* p = &xb[r * XS + 8 + q * 4];
          p[0] = v.x; p[1] = v.y; p[2] = v.z; p[3] = v.w;
        }
      } else {
        const float4* f4 = (const float4*)(fj + 32);
        #pragma unroll
        for (int q = 0; q < 8; ++q) {
          float4 v = f4[q];
          float* p = &xb[r * XS + 40 + q * 4];
          p[0] = v.x; p[1] = v.y; p[2] = v.z; p[3] = v.w;
        }
        float* p = &xb[r * XS];
        p[0] = center[3 * j + 0] - center[3 * m + 0];
        p[1] = center[3 * j + 1] - center[3 * m + 1];
        p[2] = center[3 * j + 2] - center[3 * m + 2];
        p[3] = 0.0f;
        p[4] = normal[3 * j + 0];
        p[5] = normal[3 * j + 1];
        p[6] = normal[3 * j + 2];
        p[7] = 0.0f;
      }
      __syncthreads();

      // ---------- stage A: Y_loc (K 0..3), Y_feat (K 4..71), no bias ----------
      v8f accL[4], accF[4];
      #pragma unroll
      for (int c = 0; c < 4; ++c) { accL[c] = vzero; accF[c] = vzero; }
      {
        int kk = 2 * half;
        v2f a; a.x = xb[r * XS + kk]; a.y = xb[r * XS + kk + 1];
        #pragma unroll
        for (int c = 0; c < 4; ++c) {
          int col = r + 16 * c;
          v2f b; b.x = wA_elem(w_l0, w_f0, kk, col); b.y = wA_elem(w_l0, w_f0, kk + 1, col);
          accL[c] = wmma_f32(a, b, accL[c]);
        }
      }
      for (int kb = 4; kb < 72; kb += 4) {
        int kk = kb + 2 * half;
        v2f a; a.x = xb[r * XS + kk]; a.y = xb[r * XS + kk + 1];
        #pragma unroll
        for (int c = 0; c < 4; ++c) {
          int col = r + 16 * c;
          v2f b; b.x = wA_elem(w_l0, w_f0, kk, col); b.y = wA_elem(w_l0, w_f0, kk + 1, col);
          accF[c] = wmma_f32(a, b, accF[c]);
        }
      }

      if (STAGE == 1) {
        #pragma unroll
        for (int c = 0; c < 4; ++c)
          #pragma unroll
          for (int jj = 0; jj < 8; ++jj) {
            float v = accL[c][jj]; sA[c]     += v; qA[c]     += v * v;
            float u = accF[c][jj]; sA[4 + c] += u; qA[4 + c] += u * u;
          }
      } else {
        // ---------- h0 = relu(aL*yL + aF*yF + dC) -> LDS ----------
        __syncthreads();
        #pragma unroll
        for (int c = 0; c < 4; ++c) {
          int ch = r + 16 * c;
          #pragma unroll
          for (int jj = 0; jj < 8; ++jj) {
            int row = half * 8 + jj;
            float h = fmaxf(0.0f, aL[c] * accL[c][jj] + aF[c] * accF[c][jj] + dC[c]);
            xb[row * XS + ch] = h;
          }
        }
        __syncthreads();

        // ---------- stage B: Y1 = h0 * w1^T (K = 64) ----------
        v8f acc1[4];
        #pragma unroll
        for (int c = 0; c < 4; ++c) acc1[c] = vzero;
        for (int kb = 0; kb < 64; kb += 4) {
          int kk = kb + 2 * half;
          v2f a; a.x = xb[r * XS + kk]; a.y = xb[r * XS + kk + 1];
          #pragma unroll
          for (int c = 0; c < 4; ++c) {
            int col = r + 16 * c;
            v2f b; b.x = w1[col * 64 + kk]; b.y = w1[col * 64 + kk + 1];
            acc1[c] = wmma_f32(a, b, acc1[c]);
          }
        }

        if (STAGE == 2) {
          #pragma unroll
          for (int c = 0; c < 4; ++c)
            #pragma unroll
            for (int jj = 0; jj < 8; ++jj) {
              float v = acc1[c][jj]; s1[c] += v; q1[c] += v * v;
            }
        } else {
          // ---------- h1 -> LDS ----------
          __syncthreads();
          #pragma unroll
          for (int c = 0; c < 4; ++c) {
            int ch = r + 16 * c;
            #pragma unroll
            for (int jj = 0; jj < 8; ++jj) {
              int row = half * 8 + jj;
              float h = fmaxf(0.0f, a1v[c] * acc1[c][jj] + d1v[c]);
              xb[row * XS + ch] = h;
            }
          }
          __syncthreads();

          // ---------- stage C: Y2 = h1 * w2^T (K = 64, 128 channels) ----------
          v8f acc2[8];
          #pragma unroll
          for (int c = 0; c < 8; ++c) acc2[c] = vzero;
          for (int kb = 0; kb < 64; kb += 4) {
            int kk = kb + 2 * half;
            v2f a; a.x = xb[r * XS + kk]; a.y = xb[r * XS + kk + 1];
            #pragma unroll
            for (int c = 0; c < 8; ++c) {
              int col = r + 16 * c;
              v2f b; b.x = w2[col * 64 + kk]; b.y = w2[col * 64 + kk + 1];
              acc2[c] = wmma_f32(a, b, acc2[c]);
            }
          }

          if (STAGE == 3) {
            #pragma unroll
            for (int c = 0; c < 8; ++c)
              #pragma unroll
              for (int jj = 0; jj < 8; ++jj) {
                float v = acc2[c][jj]; s2[c] += v; q2[c] += v * v;
              }
          } else { // STAGE == 4: h2 = relu(BN), running max over neighbors
            #pragma unroll
            for (int c = 0; c < 8; ++c) {
              float mx = omax[c];
              #pragma unroll
              for (int jj = 0; jj < 8; ++jj) {
                float h = fmaxf(0.0f, a2v[c] * acc2[c][jj] + d2v[c]);
                mx = fmaxf(mx, h);
              }
              omax[c] = mx;
            }
          }
        }
      }
      __syncthreads();   // protect xbuf before next tile's staging
    } // tile

    if (STAGE == 4) {
      #pragma unroll
      for (int c = 0; c < 8; ++c) {
        float o = fmaxf(omax[c], __shfl_xor(omax[c], 16, 32));
        if (half == 0) out_feat[(size_t)m * 128 + 16 * c + r] = o;
      }
    }
  } // m loop

  // ---------- flush per-lane stats: LDS reduce then global atomics ----------
  if (STAGE == 1) {
    #pragma unroll
    for (int c = 0; c < 4; ++c) {
      int ch = r + 16 * c;
      atomicAdd(&lstats[ch],        sA[c]);
      atomicAdd(&lstats[64 + ch],   qA[c]);
      atomicAdd(&lstats[128 + ch],  sA[4 + c]);
      atomicAdd(&lstats[192 + ch],  qA[4 + c]);
    }
    __syncthreads();
    for (int i = tid; i < 256; i += TPB) atomicAdd(&ws[WS_SUML + i], lstats[i]);
  } else if (STAGE == 2) {
    #pragma unroll
    for (int c = 0; c < 4; ++c) {
      int ch = r + 16 * c;
      atomicAdd(&lstats[ch],      s1[c]);
      atomicAdd(&lstats[64 + ch], q1[c]);
    }
    __syncthreads();
    for (int i = tid; i < 128; i += TPB) atomicAdd(&ws[WS_SUM1 + i], lstats[i]);
  } else if (STAGE == 3) {
    #pragma unroll
    for (int c = 0; c < 8; ++c) {
      int ch = r + 16 * c;
      atomicAdd(&lstats[ch],       s2[c]);
      atomicAdd(&lstats[128 + ch], q2[c]);
    }
    __syncthreads();
    for (int i = tid; i < 256; i += TPB) atomicAdd(&ws[WS_SUM2 + i], lstats[i]);
  }
}

__global__ void sa_zero(float* __restrict__ ws) {
  int i = blockIdx.x * blockDim.x + threadIdx.x;
  if (i < WS_NACC) ws[i] = 0.0f;
}

__global__ void sa_finalize(int stage,
    const float* __restrict__ g_l0, const float* __restrict__ be_l0,
    const float* __restrict__ g_f0, const float* __restrict__ be_f0,
    const float* __restrict__ g1,  const float* __restrict__ be1,
    const float* __restrict__ g2,  const float* __restrict__ be2,
    float* __restrict__ ws)
{
  const int tid = threadIdx.x;
  const float inv = 1.0f / (float)ROWS_TOT;
  if (stage == 1 && tid < 64) {
    float mL = ws[WS_SUML + tid] * inv;
    float vL = ws[WS_SQL  + tid] * inv - mL * mL;
    float aL = g_l0[tid] * rsqrtf(vL + 1e-5f);
    float dL = be_l0[tid] - mL * aL;
    float mF = ws[WS_SUMF + tid] * inv;
    float vF = ws[WS_SQF  + tid] * inv - mF * mF;
    float aF = g_f0[tid] * rsqrtf(vF + 1e-5f);
    float dF = be_f0[tid] - mF * aF;
    ws[WS_AL + tid] = aL; ws[WS_AF + tid] = aF; ws[WS_DC + tid] = dL + dF;
  } else if (stage == 2 && tid < 64) {
    float m1 = ws[WS_SUM1 + tid] * inv;
    float v1 = ws[WS_SQ1  + tid] * inv - m1 * m1;
    float a1 = g1[tid] * rsqrtf(v1 + 1e-5f);
    ws[WS_A1 + tid] = a1;
    ws[WS_D1 + tid] = be1[tid] - m1 * a1;
  } else if (stage == 3 && tid < 128) {
    float m2 = ws[WS_SUM2 + tid] * inv;
    float v2 = ws[WS_SQ2  + tid] * inv - m2 * m2;
    float a2 = g2[tid] * rsqrtf(v2 + 1e-5f);
    ws[WS_A2 + tid] = a2;
    ws[WS_D2 + tid] = be2[tid] - m2 * a2;
  }
}

__global__ void sa_copy(const float* __restrict__ center,
                        const float* __restrict__ normal,
                        const int* __restrict__ offset,
                        float* __restrict__ out)
{
  int i = blockIdx.x * blockDim.x + threadIdx.x;
  if (i < NPTS * 3) {
    out[i] = center[i];
  } else if (i < NPTS * 6) {
    out[i] = normal[i - NPTS * 3];
  } else if (i == NPTS * 6) {
    // offset (int32) goes last, after new_feature [NPTS,128]
    ((int*)out)[(size_t)NPTS * 6 + (size_t)NPTS * 128] = offset[0];
  }
}

extern "C" void kernel_launch(void* const* d_in, const int* in_sizes, int n_in,
                              void* d_out, int out_size, void* d_ws, size_t ws_size,
                              hipStream_t stream)
{
  (void)in_sizes; (void)n_in; (void)out_size; (void)ws_size;
  const float* center = (const float*)d_in[0];
  const float* normal = (const float*)d_in[1];
  const float* feature= (const float*)d_in[2];
  const int*   gidx   = (const int*)d_in[3];
  const int*   offset = (const int*)d_in[4];
  const float* w_l0   = (const float*)d_in[5];
  const float* g_l0   = (const float*)d_in[7];
  const float* be_l0  = (const float*)d_in[8];
  const float* w_f0   = (const float*)d_in[9];
  const float* g_f0   = (const float*)d_in[11];
  const float* be_f0  = (const float*)d_in[12];
  const float* w1     = (const float*)d_in[13];
  const float* g1     = (const float*)d_in[15];
  const float* be1    = (const float*)d_in[16];
  const float* w2     = (const float*)d_in[17];
  const float* g2     = (const float*)d_in[19];
  const float* be2    = (const float*)d_in[20];

  float* out      = (float*)d_out;
  float* ws       = (float*)d_ws;
  float* out_feat = out + (size_t)NPTS * 6;

  sa_zero<<<(WS_NACC + 255) / 256, 256, 0, stream>>>(ws);
  sa_pass<1><<<NWG, TPB, 0, stream>>>(center, normal, feature, gidx, w_l0, w_f0, w1, w2, ws, out_feat);
  sa_finalize<<<1, 128, 0, stream>>>(1, g_l0, be_l0, g_f0, be_f0, g1, be1, g2, be2, ws);
  sa_pass<2><<<NWG, TPB, 0, stream>>>(center, normal, feature, gidx, w_l0, w_f0, w1, w2, ws, out_feat);
  sa_finalize<<<1, 128, 0, stream>>>(2, g_l0, be_l0, g_f0, be_f0, g1, be1, g2, be2, ws);
  sa_pass<3><<<NWG, TPB, 0, stream>>>(center, normal, feature, gidx, w_l0, w_f0, w1, w2, ws, out_feat);
  sa_finalize<<<1, 128, 0, stream>>>(3, g_l0, be_l0, g_f0, be_f0, g1, be1, g2, be2, ws);
  sa_pass<4><<<NWG, TPB, 0, stream>>>(center, normal, feature, gidx, w_l0, w_f0, w1, w2, ws, out_feat);
  sa_copy<<<(NPTS * 6 + 1 + 255) / 256, 256, 0, stream>>>(center, normal, offset, out);
}